// UNetSphericalHealpixResidualShort3Levels_67869073211450
// MI455X (gfx1250) — compile-verified
//
#include <hip/hip_runtime.h>
#include <hip/hip_bf16.h>

// ---------------------------------------------------------------------------
// UNetSphericalHealpixResidualShort3Levels forward, fp32, WMMA f32 16x16x4.
//
// Roofline: ~83 GFLOP total, all tensors L2-resident (<=100MB each, 192MB L2)
// => bandwidth/latency bound on MI455X; fp32 WMMA keeps reference precision
// while engaging the matrix pipe.  B=32, N0=3072, K=3.
// ---------------------------------------------------------------------------

#define BB   32
#define N0   3072
#define N1   768
#define N2   192
#define M0   (BB * N0)   /* 98304 */
#define M1   (BB * N1)   /* 24576 */
#define M2   (BB * N2)   /*  6144 */
#define EPSF 1e-5f

typedef __attribute__((ext_vector_type(2))) float v2f;
typedef __attribute__((ext_vector_type(8))) float v8f;

// ---------------------------------------------------------------------------
// Sparse Laplacian apply (4 channels/thread, b128 loads):
//   out = scale * (diag[n]*x + sum_j w[n,j]*x[nbr[n,j]])  - (HAS_SUB ? sub : 0)
// x: [B,N,C] row-major, C a multiple of 16 so float4 access is aligned.
// Covers x1 = L x (scale=1) and x2 = 2 L x1 - x0 (scale=2, sub=x0).
// ---------------------------------------------------------------------------
template <int HAS_SUB>
__global__ void __launch_bounds__(256)
lap_kernel(float* __restrict__ out, const float* __restrict__ x,
           const float* __restrict__ sub,
           const int* __restrict__ nbr, const float* __restrict__ w,
           const float* __restrict__ diag, int N, int C, float scale)
{
    const int C4 = C >> 2;
    size_t i = (size_t)blockIdx.x * blockDim.x + threadIdx.x;
    size_t total = (size_t)BB * N * C4;
    if (i >= total) return;
    const int    c     = (int)(i % C4) << 2;
    const size_t bn    = i / C4;
    const int    n     = (int)(bn % N);
    const size_t bbase = (bn / N) * (size_t)N * C;

    const float dg = diag[n];
    float4 a = *(const float4*)(x + bn * C + c);
    float4 acc = make_float4(dg * a.x, dg * a.y, dg * a.z, dg * a.w);
#pragma unroll
    for (int j = 0; j < 8; ++j) {
        const int   nb = nbr[n * 8 + j];
        const float wj = w[n * 8 + j];
        float4 v = *(const float4*)(x + bbase + (size_t)nb * C + c);
        acc.x += wj * v.x; acc.y += wj * v.y;
        acc.z += wj * v.z; acc.w += wj * v.w;
    }
    float4 r = make_float4(scale * acc.x, scale * acc.y,
                           scale * acc.z, scale * acc.w);
    if (HAS_SUB) {
        float4 s = *(const float4*)(sub + bn * C + c);
        r.x -= s.x; r.y -= s.y; r.z -= s.z; r.w -= s.w;
    }
    *(float4*)(out + bn * C + c) = r;
}

// ---------------------------------------------------------------------------
// WMMA GEMM:  out[M,F] = [y0 | y1 | y2][M, nseg*C] @ W[nseg*C, F] + bias
//             (+= out if acc).  nseg = 3 for Chebyshev convs, 1 for 1x1 res.
// One wave owns a 32x16 output tile: two stacked 16x16 accumulators share the
// B fragment (halves B traffic per MAC).  V_WMMA_F32_16X16X4_F32 layouts
// (ISA 7.12.2):
//   A (16x4 f32): lane = m + 16*khalf, vgpr = k&1  -> contiguous b64 per lane
//   B (4x16 f32): lane = n + 16*khalf, vgpr = k&1  -> stride-F b32 pair
//   D (16x16):    vgpr r -> M = r + 8*lane[4], N = lane[3:0]
// Out-of-range output columns (only final layer, F=4): B index is CLAMPED,
// not predicated -- lane n's B values feed only output column n, which is
// never stored, so the hot loop carries no EXEC manipulation (WMMA requires
// EXEC == all ones anyway).  Segments are unrolled explicitly (no pointer
// array) so address-space inference keeps A/B on GLOBAL, not FLAT, loads;
// the B pointer is one running induction variable across segments since
// W's segments are contiguous.
// ---------------------------------------------------------------------------
__device__ __forceinline__ const float*
gemm_seg(v8f& c0, v8f& c1,
         const float* __restrict__ A0, const float* __restrict__ A1,
         const float* __restrict__ pB, int C, size_t sF)
{
    for (int k0 = 0; k0 < C; k0 += 16) {           // C is a multiple of 16
        __builtin_prefetch(A0 + k0 + 64, 0, 3);    // near-cache streaming
        __builtin_prefetch(pB + 16 * sF, 0, 3);
#pragma unroll
        for (int kk = 0; kk < 16; kk += 4) {
            v2f a0 = *(const v2f*)(A0 + k0 + kk);  // b64: K pair contiguous
            v2f a1 = *(const v2f*)(A1 + k0 + kk);
            v2f b;
            b.x = pB[0];
            b.y = pB[sF];
            pB += 4 * sF;
            c0 = __builtin_amdgcn_wmma_f32_16x16x4_f32(
                     false, a0, false, b, (short)0, c0, false, false);
            c1 = __builtin_amdgcn_wmma_f32_16x16x4_f32(
                     false, a1, false, b, (short)0, c1, false, false);
        }
    }
    return pB;   // lands exactly on the next segment's first row
}

__global__ void __launch_bounds__(128)
wmma_gemm3_kernel(float* __restrict__ out,
                  const float* __restrict__ y0, const float* __restrict__ y1,
                  const float* __restrict__ y2,
                  const float* __restrict__ W, const float* __restrict__ bias,
                  int M, int C, int nseg, int F, int acc)
{
    const int lane   = threadIdx.x & 31;
    const int wave   = threadIdx.x >> 5;
    const int tilesF = (F + 15) >> 4;
    const int tilesM = M >> 5;                     // 32-row tiles
    const int tile   = blockIdx.x * 4 + wave;
    if (tile >= tilesM * tilesF) return;           // whole wave exits together
    const int tm = tile / tilesF;
    const int tf = tile % tilesF;

    const int    half   = lane >> 4;               // K pair {0,1} vs {2,3}
    const int    l16    = lane & 15;
    const int    am0    = tm * 32 + l16;           // A row, lower 16x16 tile
    const int    am1    = am0 + 16;                // A row, upper 16x16 tile
    const int    bn     = tf * 16 + l16;           // output column
    const bool   bvalid = bn < F;
    const int    bnc    = bvalid ? bn : (F - 1);   // clamped load column
    const size_t sF     = (size_t)F;

    // per-lane A offsets (2*half folds the K-pair select into the base)
    const size_t offA0 = (size_t)am0 * C + 2 * half;
    const size_t offA1 = (size_t)am1 * C + 2 * half;

    v8f c0 = {}, c1 = {};
    const float* pB = W + (size_t)(2 * half) * sF + bnc;
    pB = gemm_seg(c0, c1, y0 + offA0, y0 + offA1, pB, C, sF);
    if (nseg > 1)
        pB = gemm_seg(c0, c1, y1 + offA0, y1 + offA1, pB, C, sF);
    if (nseg > 2)
        (void)gemm_seg(c0, c1, y2 + offA0, y2 + offA1, pB, C, sF);

    if (bvalid) {
        const float bb = bias[bn];
#pragma unroll
        for (int r = 0; r < 8; ++r) {
            const int mlo = tm * 32 + r + 8 * half;
            const int mhi = mlo + 16;
            size_t o0 = (size_t)mlo * sF + bn;
            size_t o1 = (size_t)mhi * sF + bn;
            float v0 = c0[r] + bb; if (acc) v0 += out[o0]; out[o0] = v0;
            float v1 = c1[r] + bb; if (acc) v1 += out[o1]; out[o1] = v1;
        }
    }
}

// ---------------------------------------------------------------------------
// BatchNorm (training statistics over B*N) + ReLU
// ---------------------------------------------------------------------------
__global__ void __launch_bounds__(256)
zero_kernel(float* __restrict__ p, int n)
{
    int i = blockIdx.x * blockDim.x + threadIdx.x;
    if (i < n) p[i] = 0.0f;
}

__global__ void __launch_bounds__(256)
bn_stats_kernel(const float* __restrict__ x, float* __restrict__ sum,
                float* __restrict__ sumsq, int M, int F, int rowsPerBlock)
{
    const int r0 = blockIdx.x * rowsPerBlock;
    const int r1 = min(M, r0 + rowsPerBlock);
    for (int cb = 0; cb < F; cb += blockDim.x) {
        const int c = cb + threadIdx.x;
        if (c < F) {
            float s = 0.0f, ss = 0.0f;
            for (int m = r0; m < r1; ++m) {
                float v = x[(size_t)m * F + c];
                s += v; ss += v * v;
            }
            atomicAdd(&sum[c], s);
            atomicAdd(&sumsq[c], ss);
        }
    }
}

__global__ void __launch_bounds__(256)
bn_finalize_kernel(const float* __restrict__ sum, const float* __restrict__ sumsq,
                   float* __restrict__ mean, float* __restrict__ rstd,
                   int F, float invM)
{
    int c = blockIdx.x * blockDim.x + threadIdx.x;
    if (c < F) {
        float mu  = sum[c] * invM;
        float var = sumsq[c] * invM - mu * mu;
        mean[c] = mu;
        rstd[c] = rsqrtf(var + EPSF);
    }
}

__global__ void __launch_bounds__(256)
bn_relu_kernel(float* __restrict__ x, const float* __restrict__ mean,
               const float* __restrict__ rstd, const float* __restrict__ gamma,
               const float* __restrict__ beta, size_t total, int F)
{
    size_t i = (size_t)blockIdx.x * blockDim.x + threadIdx.x;
    if (i >= total) return;
    int c = (int)(i % F);
    float v = gamma[c] * (x[i] - mean[c]) * rstd[c] + beta[c];
    x[i] = v > 0.0f ? v : 0.0f;
}

// ---------------------------------------------------------------------------
// PoolMaxHealpix (max over nested groups of 4, per channel, with argmax)
// ---------------------------------------------------------------------------
__global__ void __launch_bounds__(256)
pool_kernel(float* __restrict__ out, int* __restrict__ idx,
            const float* __restrict__ in, int Np, int C)
{
    size_t i = (size_t)blockIdx.x * blockDim.x + threadIdx.x;
    size_t total = (size_t)BB * Np * C;
    if (i >= total) return;
    int    c  = (int)(i % C);
    size_t bp = i / C;
    int    p  = (int)(bp % Np);
    size_t b  = bp / Np;
    const float* base = in + (((size_t)b * Np * 4 + (size_t)p * 4) * C + c);
    float best = base[0];
    int   bi   = 0;
#pragma unroll
    for (int j = 1; j < 4; ++j) {
        float v = base[(size_t)j * C];
        if (v > best) { best = v; bi = j; }   // first-max, matches jnp.argmax
    }
    out[i] = best;
    idx[i] = bi;
}

// Unpool into the low Cu channels of a [B, Np*4, Ctot] concat buffer.
__global__ void __launch_bounds__(256)
unpool_cat_kernel(float* __restrict__ cat, const float* __restrict__ x,
                  const int* __restrict__ idx, int Np, int Cu, int Ctot)
{
    size_t i = (size_t)blockIdx.x * blockDim.x + threadIdx.x;
    size_t total = (size_t)BB * Np * Cu;
    if (i >= total) return;
    int    c  = (int)(i % Cu);
    size_t bp = i / Cu;
    int    p  = (int)(bp % Np);
    size_t b  = bp / Np;
    float v  = x[i];
    int   ai = idx[i];
    size_t rowbase = ((size_t)b * Np * 4 + (size_t)p * 4) * Ctot + c;
#pragma unroll
    for (int j = 0; j < 4; ++j)
        cat[rowbase + (size_t)j * Ctot] = (j == ai) ? v : 0.0f;
}

// Copy skip tensor [Mrows,Cs] into channels [Cu, Cu+Cs) of cat [Mrows,Ctot].
__global__ void __launch_bounds__(256)
copy_cat_kernel(float* __restrict__ cat, const float* __restrict__ skip,
                size_t Mrows, int Cs, int Cu, int Ctot)
{
    size_t i = (size_t)blockIdx.x * blockDim.x + threadIdx.x;
    size_t total = Mrows * (size_t)Cs;
    if (i >= total) return;
    int    c = (int)(i % Cs);
    size_t m = i / Cs;
    cat[m * Ctot + Cu + c] = skip[i];
}

// ---------------------------------------------------------------------------
// Host-side orchestration
// ---------------------------------------------------------------------------
extern "C" void kernel_launch(void* const* d_in, const int* in_sizes, int n_in,
                              void* d_out, int out_size, void* d_ws, size_t ws_size,
                              hipStream_t stream)
{
    (void)in_sizes; (void)n_in; (void)out_size; (void)ws_size;

    // ---- input mapping: jax pytree (sorted-key) flattening of setup_inputs()
    auto F32 = [&](int i) { return (const float*)d_in[i]; };
    auto I32 = [&](int i) { return (const int*)d_in[i]; };
    const float* x = F32(0);
    // params (sorted): each cheb = beta,bias,gamma,theta ; lin = b,w
    // conv11 1-4 | conv13 5-8 | conv1_res 9-10 | conv21 11-14 | conv23 15-18 |
    // conv2_res 19-20 | conv31 21-24 | conv33 25-28 | conv3_res 29-30 |
    // uconv11 31-34 | uconv12 35-38 | uconv13 (bias,theta) 39-40 |
    // uconv1_res 41-42 | uconv21 43-46 | uconv22 47-50 | uconv2_res 51-52
    const int*   nbr0 = I32(53); const float* w0 = F32(54); const float* dg0 = F32(55);
    const int*   nbr1 = I32(56); const float* w1 = F32(57); const float* dg1 = F32(58);
    const int*   nbr2 = I32(59); const float* w2 = F32(60); const float* dg2 = F32(61);

    // ---- workspace carve-up (float units)
    float* ws = (float*)d_ws;
    size_t off = 0;
    auto alloc = [&](size_t n) { float* p = ws + off; off += n; return p; };
    float* Y1   = alloc((size_t)M0 * 256);   // lap temp 1 (max: cat0 expand)
    float* Y2   = alloc((size_t)M0 * 256);   // lap temp 2
    float* T    = alloc((size_t)M0 * 128);   // generic activation temp
    float* X1   = alloc((size_t)M0 * 128);   // level-0 skip
    float* X2I  = alloc((size_t)M1 * 128);   // pooled input, level 1
    float* X2   = alloc((size_t)M1 * 256);   // level-1 skip
    float* X3I  = alloc((size_t)M2 * 256);   // pooled input, level 2
    float* X3   = alloc((size_t)M2 * 256);   // bottleneck out
    float* CAT1 = alloc((size_t)M1 * 512);   // decoder concat, level 1
    float* U2   = alloc((size_t)M1 * 128);   // decoder out, level 1
    float* CAT0 = alloc((size_t)M0 * 256);   // decoder concat, level 0
    float* U1   = alloc((size_t)M0 * 64);    // decoder out, level 0
    int*   IDX1 = (int*)alloc((size_t)M1 * 128);
    int*   IDX2 = (int*)alloc((size_t)M2 * 256);
    float* ST   = alloc(2048);               // sum|sumsq|mean|rstd (512 each)

    // ---- launch helpers
    auto cheb_expand = [&](const float* a, const int* nb, const float* w,
                           const float* dg, int N, int C) {
        size_t tot = (size_t)BB * N * (C >> 2);
        lap_kernel<0><<<(unsigned)((tot + 255) / 256), 256, 0, stream>>>(
            Y1, a, nullptr, nb, w, dg, N, C, 1.0f);          // y1 = L a
        lap_kernel<1><<<(unsigned)((tot + 255) / 256), 256, 0, stream>>>(
            Y2, Y1, a, nb, w, dg, N, C, 2.0f);               // y2 = 2 L y1 - a
    };
    auto gemm3 = [&](float* o, const float* y0, const float* y1, const float* y2,
                     const float* W, const float* bias, int M, int C, int nseg,
                     int Fo, int acc) {
        int tiles = (M >> 5) * ((Fo + 15) >> 4);
        wmma_gemm3_kernel<<<(tiles + 3) / 4, 128, 0, stream>>>(
            o, y0, y1, y2, W, bias, M, C, nseg, Fo, acc);
    };
    auto bnrelu = [&](float* a, int M, int Fo, const float* g, const float* be) {
        zero_kernel<<<4, 256, 0, stream>>>(ST, 1024);
        bn_stats_kernel<<<(M + 255) / 256, 256, 0, stream>>>(a, ST, ST + 512, M, Fo, 256);
        bn_finalize_kernel<<<2, 256, 0, stream>>>(ST, ST + 512, ST + 1024, ST + 1536,
                                                  Fo, 1.0f / (float)M);
        size_t tot = (size_t)M * Fo;
        bn_relu_kernel<<<(unsigned)((tot + 255) / 256), 256, 0, stream>>>(
            a, ST + 1024, ST + 1536, g, be, tot, Fo);
    };
    // full cheb conv + BN + ReLU block
    auto block = [&](float* o, const float* a, const int* nb, const float* w,
                     const float* dg, int N, int M, int C, int Fo,
                     const float* theta, const float* bias,
                     const float* gamma, const float* beta) {
        cheb_expand(a, nb, w, dg, N, C);
        gemm3(o, a, Y1, Y2, theta, bias, M, C, 3, Fo, 0);
        bnrelu(o, M, Fo, gamma, beta);
    };
    auto res = [&](float* o, const float* a, const float* W, const float* b,
                   int M, int C, int Fo) {      // o += a @ W + b
        gemm3(o, a, nullptr, nullptr, W, b, M, C, 1, Fo, 1);
    };

    // =========================== encoder =================================
    // level 0
    block(T,  x, nbr0, w0, dg0, N0, M0, 16,  64, F32(4), F32(2), F32(3), F32(1)); // conv11
    block(X1, T, nbr0, w0, dg0, N0, M0, 64, 128, F32(8), F32(6), F32(7), F32(5)); // conv13
    res(X1, x, F32(10), F32(9), M0, 16, 128);                                     // conv1_res
    {   size_t tot = (size_t)BB * N1 * 128;
        pool_kernel<<<(unsigned)((tot + 255) / 256), 256, 0, stream>>>(X2I, IDX1, X1, N1, 128); }
    // level 1
    block(T,  X2I, nbr1, w1, dg1, N1, M1, 128, 192, F32(14), F32(12), F32(13), F32(11)); // conv21
    block(X2, T,   nbr1, w1, dg1, N1, M1, 192, 256, F32(18), F32(16), F32(17), F32(15)); // conv23
    res(X2, X2I, F32(20), F32(19), M1, 128, 256);                                        // conv2_res
    {   size_t tot = (size_t)BB * N2 * 256;
        pool_kernel<<<(unsigned)((tot + 255) / 256), 256, 0, stream>>>(X3I, IDX2, X2, N2, 256); }
    // level 2 (bottleneck)
    block(T,  X3I, nbr2, w2, dg2, N2, M2, 256, 512, F32(24), F32(22), F32(23), F32(21)); // conv31
    block(X3, T,   nbr2, w2, dg2, N2, M2, 512, 256, F32(28), F32(26), F32(27), F32(25)); // conv33
    res(X3, X3I, F32(30), F32(29), M2, 256, 256);                                        // conv3_res

    // =========================== decoder =================================
    // unpool level2 -> level1, concat with X2
    {   size_t tot = (size_t)BB * N2 * 256;
        unpool_cat_kernel<<<(unsigned)((tot + 255) / 256), 256, 0, stream>>>(CAT1, X3, IDX2, N2, 256, 512);
        size_t tot2 = (size_t)M1 * 256;
        copy_cat_kernel<<<(unsigned)((tot2 + 255) / 256), 256, 0, stream>>>(CAT1, X2, (size_t)M1, 256, 256, 512); }
    block(T,  CAT1, nbr1, w1, dg1, N1, M1, 512, 256, F32(46), F32(44), F32(45), F32(43)); // uconv21
    block(U2, T,    nbr1, w1, dg1, N1, M1, 256, 128, F32(50), F32(48), F32(49), F32(47)); // uconv22
    res(U2, CAT1, F32(52), F32(51), M1, 512, 128);                                        // uconv2_res
    // unpool level1 -> level0, concat with X1
    {   size_t tot = (size_t)BB * N1 * 128;
        unpool_cat_kernel<<<(unsigned)((tot + 255) / 256), 256, 0, stream>>>(CAT0, U2, IDX1, N1, 128, 256);
        size_t tot2 = (size_t)M0 * 128;
        copy_cat_kernel<<<(unsigned)((tot2 + 255) / 256), 256, 0, stream>>>(CAT0, X1, (size_t)M0, 128, 128, 256); }
    block(T,  CAT0, nbr0, w0, dg0, N0, M0, 256, 128, F32(34), F32(32), F32(33), F32(31)); // uconv11
    block(U1, T,    nbr0, w0, dg0, N0, M0, 128,  64, F32(38), F32(36), F32(37), F32(35)); // uconv12
    res(U1, CAT0, F32(42), F32(41), M0, 256, 64);                                         // uconv1_res
    // final cheb conv, no BN/ReLU, straight to d_out
    cheb_expand(U1, nbr0, w0, dg0, N0, 64);
    gemm3((float*)d_out, U1, Y1, Y2, F32(40), F32(39), M0, 64, 3, 4, 0);                  // uconv13
}